// MoCo_4028679324253
// MI455X (gfx1250) — compile-verified
//
#include <hip/hip_runtime.h>
#include <hip/hip_bf16.h>

#define BATCH      256
#define IN_DIM     2048
#define DIM        128
#define KQ         65536
#define N_HARD     32
#define S1N        16
#define S2N        16
#define ROW_STRIDE (1 + KQ + S1N + S2N)   /* 65569 */
#define INV_T      (1.0f / 0.07f)

typedef float v2f __attribute__((ext_vector_type(2)));
typedef float v8f __attribute__((ext_vector_type(8)));

// ---------------- block-wide sum (all threads get result) ----------------
__device__ __forceinline__ float blk_sum(float v, float* scr) {
    #pragma unroll
    for (int o = 16; o > 0; o >>= 1) v += __shfl_xor(v, o, 32);
    const int lane = threadIdx.x & 31;
    const int wid  = threadIdx.x >> 5;
    const int nw   = blockDim.x >> 5;
    if (lane == 0) scr[wid] = v;
    __syncthreads();
    float s = 0.f;
    for (int i = 0; i < nw; ++i) s += scr[i];
    __syncthreads();
    return s;
}

// ---------------- kernel 1: encoders + pos logit + labels ----------------
// block = 128 threads (thread d = output dim), grid = 256 rows
__global__ __launch_bounds__(128) void k_encode(
    const float* __restrict__ q, const float* __restrict__ k,
    const float* __restrict__ Wq, const float* __restrict__ Wk,
    float* __restrict__ out_q, float* __restrict__ out) {
    __shared__ float lq[IN_DIM];
    __shared__ float lk[IN_DIM];
    __shared__ float scr[4];
    const int b = blockIdx.x, d = threadIdx.x;
    for (int i = d; i < IN_DIM; i += 128) {
        lq[i] = q[b * IN_DIM + i];
        lk[i] = k[b * IN_DIM + i];
    }
    __syncthreads();
    float aq = 0.f, ak = 0.f;
    #pragma unroll 8
    for (int i = 0; i < IN_DIM; ++i) {
        const float wq = Wq[i * DIM + d];
        const float wk = Wk[i * DIM + d];
        aq = fmaf(lq[i], wq, aq);
        ak = fmaf(lk[i], fmaf(0.999f, wk, 0.001f * wq), ak);  // W_k_new fused
    }
    const float nq2 = blk_sum(aq * aq, scr);
    const float nk2 = blk_sum(ak * ak, scr);
    const float dp  = blk_sum(aq * ak, scr);
    const float nq = fmaxf(sqrtf(nq2), 1e-12f);
    const float nk = fmaxf(sqrtf(nk2), 1e-12f);
    out_q[b * DIM + d] = aq / nq;
    if (d == 0) {
        out[(size_t)b * ROW_STRIDE] = (dp / (nq * nk)) * INV_T;   // logits_pos
        out[(size_t)BATCH * ROW_STRIDE + b] = 0.0f;               // labels (int 0 bits)
    }
}

// ---------------- kernel 2: logits_neg = out_q @ queue^T / T  (f32 WMMA) ----
// block = 256 threads = 8 waves; wave w -> m_tile = w + 8*blockIdx.y (grid.y=2)
// blockIdx.x = n_tile (KQ/16 = 4096). 16x16 C tile, K=128 in steps of 4.
__global__ __launch_bounds__(256) void k_neg_gemm(
    const float* __restrict__ out_q, const float* __restrict__ queue,
    float* __restrict__ out) {
    const int lane  = threadIdx.x & 31;
    const int wave  = threadIdx.x >> 5;
    const int m0    = (wave + 8 * blockIdx.y) * 16;
    const int n0    = blockIdx.x * 16;
    const int row16 = lane & 15;         // M for A-frag / N for B-frag
    const int kbase = (lane >> 4) * 2;   // K pair selector per §7.12.2
    const float* aptr = out_q + (size_t)(m0 + row16) * DIM + kbase;
    const float* bptr = queue + (size_t)(n0 + row16) * DIM + kbase;
    v8f acc = {};
    #pragma unroll 4
    for (int k0 = 0; k0 < DIM; k0 += 4) {
        v2f a = *(const v2f*)(aptr + k0);
        v2f bb = *(const v2f*)(bptr + k0);
        acc = __builtin_amdgcn_wmma_f32_16x16x4_f32(
            false, a, false, bb, (short)0, acc, false, false);
    }
    // C layout: VGPR r -> M = r + 8*(lane>>4), N = lane&15
    const int mhi = (lane >> 4) * 8;
    const size_t col = 1 + (size_t)n0 + (lane & 15);
    #pragma unroll
    for (int r = 0; r < 8; ++r) {
        const int m = m0 + r + mhi;
        out[(size_t)m * ROW_STRIDE + col] = acc[r] * INV_T;
    }
}

// ---------------- kernel 3: sorted top-32 indices per row ----------------
// block = 256 threads, grid = 256 rows
__global__ __launch_bounds__(256) void k_topk(
    const float* __restrict__ out, int* __restrict__ hard) {
    __shared__ unsigned long long keys[256 * N_HARD];  // 64 KB
    __shared__ unsigned long long scr[8];
    __shared__ unsigned long long bmax;
    const int b = blockIdx.x, t = threadIdx.x;
    const float* row = out + (size_t)b * ROW_STRIDE + 1;

    float sv[N_HARD]; int si[N_HARD];
    #pragma unroll
    for (int r = 0; r < N_HARD; ++r) { sv[r] = -3.4e38f; si[r] = 0; }

    for (int j = t; j < KQ; j += 256) {
        const float v = row[j];
        if (v > sv[N_HARD - 1]) {            // branchless unrolled insertion
            float pv = v; int pi = j;
            #pragma unroll
            for (int r = 0; r < N_HARD; ++r) {
                const bool c = pv > sv[r];   // ties keep earlier index first
                const float tv = sv[r]; const int ti = si[r];
                sv[r] = c ? pv : sv[r];  si[r] = c ? pi : si[r];
                pv = c ? tv : pv;        pi = c ? ti : pi;
            }
        }
    }
    // order-preserving u64 keys: (flipped float bits << 32) | ~idx
    #pragma unroll
    for (int r = 0; r < N_HARD; ++r) {
        unsigned int fb = __float_as_uint(sv[r]);
        fb ^= ((int)fb < 0) ? 0xFFFFFFFFu : 0x80000000u;
        keys[t * N_HARD + r] =
            ((unsigned long long)fb << 32) | (unsigned int)(~si[r]);
    }
    __syncthreads();

    int p = 0;
    for (int r = 0; r < N_HARD; ++r) {
        unsigned long long key = (p < N_HARD) ? keys[t * N_HARD + p] : 0ull;
        unsigned long long kk = key;
        #pragma unroll
        for (int o = 16; o > 0; o >>= 1) {
            unsigned long long other = __shfl_xor(kk, o, 32);
            kk = (kk > other) ? kk : other;
        }
        if ((t & 31) == 0) scr[t >> 5] = kk;
        __syncthreads();
        if (t == 0) {
            unsigned long long m = 0;
            for (int i = 0; i < 8; ++i) m = (m > scr[i]) ? m : scr[i];
            bmax = m;
        }
        __syncthreads();
        const unsigned long long m = bmax;
        if (key == m && m != 0) ++p;                       // winner advances
        if (t == 0) hard[b * N_HARD + r] = (int)(~(unsigned int)m);
        __syncthreads();
    }
}

// ---------------- kernel 4: mixed hard negatives (type1 + type2) ----------
// block = 128 threads (thread d = dim), grid = 256 rows
__global__ __launch_bounds__(128) void k_mix(
    const float* __restrict__ out_q, const float* __restrict__ queue,
    const int* __restrict__ hard,
    const float* __restrict__ alpha, const float* __restrict__ beta,
    const int* __restrict__ i1a, const int* __restrict__ i1b,
    const int* __restrict__ i2, const int* __restrict__ stepp,
    float* __restrict__ out) {
    __shared__ int h[N_HARD];
    __shared__ float scr[4];
    const int b = blockIdx.x, d = threadIdx.x;
    const int step = stepp[0];
    if (d < N_HARD) h[d] = hard[b * N_HARD + d];
    __syncthreads();
    const float oq = out_q[b * DIM + d];

    if (step > 1024) {
        for (int s = 0; s < S1N; ++s) {
            const int g1 = h[i1a[b * S1N + s] & (N_HARD - 1)];
            const int g2 = h[i1b[b * S1N + s] & (N_HARD - 1)];
            const float a = alpha[b * S1N + s];
            const float m = a * queue[(size_t)g1 * DIM + d] +
                            (1.f - a) * queue[(size_t)g2 * DIM + d];
            const float ss = blk_sum(m * m, scr);
            const float dp = blk_sum(oq * m, scr);
            if (d == 0)
                out[(size_t)b * ROW_STRIDE + 1 + KQ + s] =
                    dp / fmaxf(sqrtf(ss), 1e-12f) * INV_T;
        }
    }
    if (step > 2048) {
        for (int s = 0; s < S2N; ++s) {
            const int g = h[i2[b * S2N + s] & (N_HARD - 1)];
            const float bf = beta[b * S2N + s] * 0.5f;
            const float m = bf * oq + (1.f - bf) * queue[(size_t)g * DIM + d];
            const float ss = blk_sum(m * m, scr);
            const float dp = blk_sum(oq * m, scr);
            if (d == 0)
                out[(size_t)b * ROW_STRIDE + 1 + KQ + S1N + s] =
                    dp / fmaxf(sqrtf(ss), 1e-12f) * INV_T;
        }
    }
}

extern "C" void kernel_launch(void* const* d_in, const int* in_sizes, int n_in,
                              void* d_out, int out_size, void* d_ws, size_t ws_size,
                              hipStream_t stream) {
    const float* q     = (const float*)d_in[0];
    const float* k     = (const float*)d_in[1];
    const float* Wq    = (const float*)d_in[2];
    const float* Wk    = (const float*)d_in[3];
    const float* queue = (const float*)d_in[4];
    const float* alpha = (const float*)d_in[5];
    const float* beta  = (const float*)d_in[6];
    const int*   i1a   = (const int*)d_in[7];
    const int*   i1b   = (const int*)d_in[8];
    const int*   i2    = (const int*)d_in[9];
    const int*   step  = (const int*)d_in[10];
    float* out = (float*)d_out;

    float* out_q = (float*)d_ws;                                  // 256*128 f32
    int*   hard  = (int*)((char*)d_ws + BATCH * DIM * sizeof(float)); // 256*32 i32

    k_encode<<<BATCH, 128, 0, stream>>>(q, k, Wq, Wk, out_q, out);
    dim3 g2(KQ / 16, 2, 1);
    k_neg_gemm<<<g2, 256, 0, stream>>>(out_q, queue, out);
    k_topk<<<BATCH, 256, 0, stream>>>(out, hard);
    k_mix<<<BATCH, 128, 0, stream>>>(out_q, queue, hard, alpha, beta,
                                     i1a, i1b, i2, step, out);
}